// ChamferDistance_16003048145306
// MI455X (gfx1250) — compile-verified
//
#include <hip/hip_runtime.h>
#include <math.h>

// Chamfer distance, B=4, N=M=8192, D=3, fp32.
// cross(a,b) terms computed with V_WMMA_F32_16X16X4_F32 (K padded 3->4).
// Both min-directions fused into one launch for 2x wave parallelism.

typedef float v2f __attribute__((ext_vector_type(2)));
typedef float v8f __attribute__((ext_vector_type(8)));

#define WAVES_PER_BLOCK 8
#define BLOCK_THREADS   256

// For each a-point: min over all b-points of (||b||^2 - 2 a.b).
// (row-constant ||a||^2 is added back in the reduction kernel.)
// A-fragment (ISA 7.12.2, 32-bit A 16x4): lane%16 = row M; lanes 0-15 hold
// K0,K1; lanes 16-31 hold K2,K3. B-fragment mirrored (lane%16 = col N).
// C/D layout: VGPR v, lanes 0-15 -> (M=v, N=lane); lanes 16-31 -> (M=v+8).
__global__ __launch_bounds__(BLOCK_THREADS)
void chamfer_min_kernel(const float* __restrict__ pred,
                        const float* __restrict__ targ,
                        float* __restrict__ minPred,
                        float* __restrict__ minTarg,
                        int N, int M, int blocksDir0)
{
    // Fused directions: first blocksDir0 blocks do pred-vs-targ, rest swapped.
    const float* Apts;
    const float* Bpts;
    float* outMin;
    int NA, NB;
    int blk = blockIdx.x;
    if (blk < blocksDir0) {
        Apts = pred; Bpts = targ; outMin = minPred; NA = N; NB = M;
    } else {
        blk -= blocksDir0;
        Apts = targ; Bpts = pred; outMin = minTarg; NA = M; NB = N;
    }

    const int lane = threadIdx.x & 31;
    const int half = lane >> 4;   // 0: K0/K1 pair, 1: K2/K3 pair
    const int idx  = lane & 15;   // row (A) / col (B) index within tile

    const int waveId        = blk * WAVES_PER_BLOCK + (threadIdx.x >> 5);
    const int tilesPerBatch = NA >> 4;
    const int batch         = waveId / tilesPerBatch;
    const int tile          = waveId % tilesPerBatch;

    // ---- A matrix: 16 a-points x K=4 (K=3 real, pad 0) ----
    const float* ap = Apts + ((size_t)batch * NA + (size_t)tile * 16 + idx) * 3;
    const float ax = ap[0], ay = ap[1], az = ap[2];
    v2f a;
    a.x = half ? az  : ax;   // K0 (lanes 0-15) / K2 (lanes 16-31)
    a.y = half ? 0.f : ay;   // K1 (lanes 0-15) / K3=0 (lanes 16-31)

    const float* bbase = Bpts + (size_t)batch * NB * 3;

    float acc[8];
#pragma unroll
    for (int v = 0; v < 8; ++v) acc[v] = 3.402823466e38f;

    for (int t = 0; t < NB; t += 16) {
        const float* bp = bbase + (size_t)(t + idx) * 3;
        const float bx = bp[0], by = bp[1], bz = bp[2];
        v2f b;
        b.x = half ? bz  : bx;
        b.y = half ? 0.f : by;
        const float tn = fmaf(bx, bx, fmaf(by, by, bz * bz)); // ||b||^2

        v8f c = {};
        // D = A x B (+0): 256 3-element dot products in one instruction.
        c = __builtin_amdgcn_wmma_f32_16x16x4_f32(
                /*neg_a=*/false, a, /*neg_b=*/false, b,
                /*c_mod=*/(short)0, c, /*reuse_a=*/false, /*reuse_b=*/false);

#pragma unroll
        for (int v = 0; v < 8; ++v) {
            // candidate = ||b||^2 - 2 * (a . b)
            acc[v] = fminf(acc[v], fmaf(c[v], -2.0f, tn));
        }
    }

    // min across the 16 lanes of each half (xor masks 1,2,4,8 stay in-half)
#pragma unroll
    for (int v = 0; v < 8; ++v) {
        float m = acc[v];
        m = fminf(m, __shfl_xor(m, 1, 32));
        m = fminf(m, __shfl_xor(m, 2, 32));
        m = fminf(m, __shfl_xor(m, 4, 32));
        m = fminf(m, __shfl_xor(m, 8, 32));
        acc[v] = m;
    }

    if (idx == 0) {  // lane 0 writes rows v, lane 16 writes rows v+8
        float* o = outMin + (size_t)batch * NA + (size_t)tile * 16 + half * 8;
#pragma unroll
        for (int v = 0; v < 8; ++v) o[v] = acc[v];
    }
}

// Per-element: d = sqrt(max(min_partial + ||a||^2, 0)) * w ; block-sum -> partials
__global__ __launch_bounds__(BLOCK_THREADS)
void chamfer_partial_kernel(const float* __restrict__ pts,
                            const float* __restrict__ mins,
                            int n, float w,
                            float* __restrict__ partials)
{
    __shared__ float sdata[BLOCK_THREADS];
    const int i = blockIdx.x * BLOCK_THREADS + threadIdx.x;
    float v = 0.f;
    if (i < n) {
        const float* p = pts + (size_t)i * 3;
        const float pn = fmaf(p[0], p[0], fmaf(p[1], p[1], p[2] * p[2]));
        const float d2 = fmaxf(mins[i] + pn, 0.f);
        v = sqrtf(d2) * w;
    }
    sdata[threadIdx.x] = v;
    __syncthreads();
    for (int s = BLOCK_THREADS / 2; s > 0; s >>= 1) {
        if (threadIdx.x < s) sdata[threadIdx.x] += sdata[threadIdx.x + s];
        __syncthreads();
    }
    if (threadIdx.x == 0) partials[blockIdx.x] = sdata[0];
}

// Single-block deterministic final sum.
__global__ __launch_bounds__(BLOCK_THREADS)
void chamfer_final_kernel(const float* __restrict__ partials, int n,
                          float* __restrict__ out)
{
    __shared__ float sdata[BLOCK_THREADS];
    float v = 0.f;
    for (int i = threadIdx.x; i < n; i += BLOCK_THREADS) v += partials[i];
    sdata[threadIdx.x] = v;
    __syncthreads();
    for (int s = BLOCK_THREADS / 2; s > 0; s >>= 1) {
        if (threadIdx.x < s) sdata[threadIdx.x] += sdata[threadIdx.x + s];
        __syncthreads();
    }
    if (threadIdx.x == 0) out[0] = sdata[0];
}

extern "C" void kernel_launch(void* const* d_in, const int* in_sizes, int n_in,
                              void* d_out, int out_size, void* d_ws, size_t ws_size,
                              hipStream_t stream)
{
    (void)in_sizes; (void)n_in; (void)out_size; (void)ws_size;

    const float* pred = (const float*)d_in[0];   // (B, N, 3)
    const float* targ = (const float*)d_in[1];   // (B, M, 3)
    float* out = (float*)d_out;

    const int B = 4, N = 8192, M = 8192;

    float* minPred  = (float*)d_ws;               // B*N: min over targets per pred
    float* minTarg  = minPred + (size_t)B * N;    // B*M: min over preds per target
    float* partials = minTarg + (size_t)B * M;

    // Both directions in a single launch (independent halves of the grid).
    const int blocksDir0 = (B * (N / 16)) / WAVES_PER_BLOCK;   // 256
    const int blocksDir1 = (B * (M / 16)) / WAVES_PER_BLOCK;   // 256
    chamfer_min_kernel<<<blocksDir0 + blocksDir1, BLOCK_THREADS, 0, stream>>>(
        pred, targ, minPred, minTarg, N, M, blocksDir0);

    const int g1 = (B * N + BLOCK_THREADS - 1) / BLOCK_THREADS;
    const int g2 = (B * M + BLOCK_THREADS - 1) / BLOCK_THREADS;
    const float w1 = 0.5f / ((float)N * (float)B);
    const float w2 = 0.5f / ((float)M * (float)B);

    chamfer_partial_kernel<<<g1, BLOCK_THREADS, 0, stream>>>(pred, minPred, B * N, w1, partials);
    chamfer_partial_kernel<<<g2, BLOCK_THREADS, 0, stream>>>(targ, minTarg, B * M, w2, partials + g1);
    chamfer_final_kernel<<<1, BLOCK_THREADS, 0, stream>>>(partials, g1 + g2, out);
}